// DiffNet_40853728919738
// MI455X (gfx1250) — compile-verified
//
#include <hip/hip_runtime.h>
#include <math.h>

#define NU 50000
#define NI 50000
#define DD 128
#define RD 256
#define NS 800000
#define NC 800000
#define NB 16384
#define NBLK_RED 1024

// padded LDS row stride (floats) for the 16x256 A tile: stride 260 -> bank
// step of 4 per row, conflict-free for the 16-lane b64 read pattern.
#define APAD 260

typedef __attribute__((ext_vector_type(2))) float v2f;
typedef __attribute__((ext_vector_type(8))) float v8f;

// ---------------- stage-1 reduction: per-block sum / sumsq ----------------
__global__ void reduce_partial(const float* __restrict__ x, long long n,
                               float* __restrict__ part) {
  __shared__ float ssum[256];
  __shared__ float ssq[256];
  float s = 0.f, q = 0.f;
  for (long long i = (long long)blockIdx.x * 256 + threadIdx.x; i < n;
       i += (long long)gridDim.x * 256) {
    float v = x[i];
    s += v;
    q += v * v;
  }
  ssum[threadIdx.x] = s;
  ssq[threadIdx.x] = q;
  __syncthreads();
  for (int o = 128; o > 0; o >>= 1) {
    if ((int)threadIdx.x < o) {
      ssum[threadIdx.x] += ssum[threadIdx.x + o];
      ssq[threadIdx.x] += ssq[threadIdx.x + o];
    }
    __syncthreads();
  }
  if (threadIdx.x == 0) {
    part[2 * blockIdx.x + 0] = ssum[0];
    part[2 * blockIdx.x + 1] = ssq[0];
  }
}

// ---------------- stage-2 reduction: mean and 0.2/(std+1e-9) ----------------
__global__ void reduce_final(const float* __restrict__ part, long long n,
                             float* __restrict__ stats) {
  __shared__ float ssum[256];
  __shared__ float ssq[256];
  float s = 0.f, q = 0.f;
  for (int i = threadIdx.x; i < NBLK_RED; i += 256) {
    s += part[2 * i + 0];
    q += part[2 * i + 1];
  }
  ssum[threadIdx.x] = s;
  ssq[threadIdx.x] = q;
  __syncthreads();
  for (int o = 128; o > 0; o >>= 1) {
    if ((int)threadIdx.x < o) {
      ssum[threadIdx.x] += ssum[threadIdx.x + o];
      ssq[threadIdx.x] += ssq[threadIdx.x + o];
    }
    __syncthreads();
  }
  if (threadIdx.x == 0) {
    double m = (double)ssum[0] / (double)n;
    double var = ((double)ssq[0] - (double)n * m * m) / (double)(n - 1);
    if (var < 0.0) var = 0.0;
    float sd = sqrtf((float)var) + 1e-9f;
    stats[0] = (float)m;
    stats[1] = 0.2f / sd;
  }
}

// ---------------- FP32 WMMA GEMM: P = norm(X) @ W + bias ----------------
// X: [Nrows, 256] normalized on the fly with (x - stats[0]) * stats[1]
// W: [256, 128], bias: [128], P: [Nrows, 128]
// grid.x = Nrows/16 (16-row strip per block); 8 waves -> 8 column tiles.
// The 16x256 A strip is staged once per block into LDS with
// GLOBAL_LOAD_ASYNC_TO_LDS_B128 (ASYNCcnt path), then each wave feeds its
// WMMA A operand from LDS (conflict-free padded layout).
__global__ void gemm_norm_wmma(const float* __restrict__ X,
                               const float* __restrict__ stats,
                               const float* __restrict__ W,
                               const float* __restrict__ bias,
                               float* __restrict__ P) {
  __shared__ __align__(16) float tileA[16 * APAD];

  const int lane = threadIdx.x & 31;
  const int wave = threadIdx.x >> 5;
  const int row0 = blockIdx.x * 16;
  const int col0 = wave * 16;

  // ---- async-stage the 16x256 f32 strip (16 KB) into LDS ----
  // packed chunk c (16 B) covers floats [c*4, c*4+3]; row = c>>6, col = (c&63)*4
  {
    const unsigned long long gbase =
        (unsigned long long)(const void*)(X + (long long)row0 * RD);
    const unsigned lds_base = (unsigned)(unsigned long long)(void*)tileA;
#pragma unroll
    for (int j = 0; j < 4; ++j) {
      const int c = (int)threadIdx.x + 256 * j;
      const unsigned long long ga = gbase + (unsigned)(c * 16);
      const unsigned la =
          lds_base + (unsigned)((c >> 6) * (APAD * 4) + (c & 63) * 16);
      asm volatile("global_load_async_to_lds_b128 %0, %1, off"
                   :
                   : "v"(la), "v"(ga)
                   : "memory");
    }
    asm volatile("s_wait_asynccnt 0" ::: "memory");
  }
  __syncthreads();

  const float m = stats[0];
  const float sc = stats[1];

  // A 16x4 f32 layout: lanes 0-15 carry M=lane, K={k,k+1}; lanes 16-31 carry
  // M=lane-16, K={k+2,k+3}.  B 4x16 symmetric split across lane halves.
  const int r = lane & 15;
  const int kh = (lane >> 4) << 1;  // 0 or 2

  v8f c = {0.f, 0.f, 0.f, 0.f, 0.f, 0.f, 0.f, 0.f};
  const float* arow = &tileA[r * APAD];
  const float* wcol = W + col0 + r;

#pragma unroll 4
  for (int k = 0; k < RD; k += 4) {
    const int kk = k + kh;
    float2 av = *(const float2*)(arow + kk);  // ds_load_b64, conflict-free
    v2f a;
    a.x = (av.x - m) * sc;
    a.y = (av.y - m) * sc;
    v2f b;
    b.x = wcol[kk * DD];
    b.y = wcol[(kk + 1) * DD];
    c = __builtin_amdgcn_wmma_f32_16x16x4_f32(
        /*neg_a=*/false, a, /*neg_b=*/false, b,
        /*c_mod=*/(short)0, c, /*reuse_a=*/false, /*reuse_b=*/false);
  }

  // C/D 16x16 f32 layout: VGPR v -> M = v + 8*(lane>=16), N = lane%16.
  const int n = lane & 15;
  const int mbase = (lane >> 4) << 3;
  const float bb = bias[col0 + n];
#pragma unroll
  for (int v = 0; v < 8; ++v) {
    P[(long long)(row0 + mbase + v) * DD + col0 + n] = c[v] + bb;
  }
}

// ---------------- fuse: embedding + norm(P) ----------------
__global__ void fuse_kernel(const float* __restrict__ uemb,
                            const float* __restrict__ iemb,
                            const float* __restrict__ Pu,
                            const float* __restrict__ Pi,
                            const float* __restrict__ stats4,  // [mu,su,mi,si]
                            float* __restrict__ fusion_user,
                            float* __restrict__ final_item) {
  const float mu = stats4[0], su = stats4[1];
  const float mi = stats4[2], si = stats4[3];
  const long long n = (long long)NU * DD;
  for (long long i = (long long)blockIdx.x * 256 + threadIdx.x; i < n;
       i += (long long)gridDim.x * 256) {
    fusion_user[i] = uemb[i] + (Pu[i] - mu) * su;
    final_item[i] = iemb[i] + (Pi[i] - mi) * si;
  }
}

__global__ void zero_kernel(float* __restrict__ p, long long n) {
  for (long long i = (long long)blockIdx.x * 256 + threadIdx.x; i < n;
       i += (long long)gridDim.x * 256)
    p[i] = 0.f;
}

// ---------------- COO SpMM via float atomics: out[row] += v * X[col] -------
__global__ void spmm_atomic(const int* __restrict__ idx,
                            const float* __restrict__ vals, int nnz,
                            const float* __restrict__ X,
                            float* __restrict__ out) {
  const int e = blockIdx.x * 8 + (threadIdx.x >> 5);
  const int lane = threadIdx.x & 31;
  if (e >= nnz) return;
  const int row = idx[2 * e + 0];
  const int col = idx[2 * e + 1];
  const float v = vals[e];
  const float* xp = X + (long long)col * DD + lane * 4;
  __builtin_prefetch(xp, 0, 0);
  const float4 xv = *(const float4*)xp;
  float* op = out + (long long)row * DD + lane * 4;
  atomicAdd(op + 0, v * xv.x);
  atomicAdd(op + 1, v * xv.y);
  atomicAdd(op + 2, v * xv.z);
  atomicAdd(op + 3, v * xv.w);
}

// ---------------- final gather + dot + sigmoid -----------------------------
__global__ void score_kernel(const int* __restrict__ uid,
                             const int* __restrict__ iid,
                             const float* __restrict__ us,
                             const float* __restrict__ ui,
                             const float* __restrict__ u2,
                             const float* __restrict__ fit,
                             float* __restrict__ out) {
  const int b = blockIdx.x * 8 + (threadIdx.x >> 5);
  const int lane = threadIdx.x & 31;
  if (b >= NB) return;
  const long long u = uid[b];
  const long long it = iid[b];
  const float4 a0 = *(const float4*)(us + u * DD + lane * 4);
  const float4 a1 = *(const float4*)(ui + u * DD + lane * 4);
  const float4 a2 = *(const float4*)(u2 + u * DD + lane * 4);
  const float4 w = *(const float4*)(fit + it * DD + lane * 4);
  float s = (a0.x + a1.x + a2.x) * w.x + (a0.y + a1.y + a2.y) * w.y +
            (a0.z + a1.z + a2.z) * w.z + (a0.w + a1.w + a2.w) * w.w;
  for (int o = 16; o > 0; o >>= 1) s += __shfl_xor(s, o);
  if (lane == 0) out[b] = 1.f / (1.f + expf(-s));
}

extern "C" void kernel_launch(void* const* d_in, const int* in_sizes, int n_in,
                              void* d_out, int out_size, void* d_ws,
                              size_t ws_size, hipStream_t stream) {
  (void)in_sizes; (void)n_in; (void)out_size; (void)ws_size;
  const int* user_ids = (const int*)d_in[0];
  const int* item_ids = (const int*)d_in[1];
  const int* soc_idx = (const int*)d_in[2];
  const float* soc_val = (const float*)d_in[3];
  const int* con_idx = (const int*)d_in[4];
  const float* con_val = (const float*)d_in[5];
  const float* user_emb = (const float*)d_in[6];
  const float* item_emb = (const float*)d_in[7];
  const float* user_rev = (const float*)d_in[8];
  const float* item_rev = (const float*)d_in[9];
  const float* W = (const float*)d_in[10];
  const float* bias = (const float*)d_in[11];
  float* out = (float*)d_out;

  const long long ED = (long long)NU * DD;  // 6.4M floats per matrix
  float* w = (float*)d_ws;
  float* P_user = w;
  float* P_item = P_user + ED;
  float* fusion_user = P_item + ED;
  float* final_item = fusion_user + ED;
  float* u_soc = final_item + ED;
  float* u_items = u_soc + ED;
  float* u_soc2 = u_items + ED;
  float* part = u_soc2 + ED;            // 2*NBLK_RED floats
  float* stats = part + 2 * NBLK_RED;   // 8 floats

  const long long nrev = (long long)NU * RD;

  // stats for user_review / item_review
  reduce_partial<<<NBLK_RED, 256, 0, stream>>>(user_rev, nrev, part);
  reduce_final<<<1, 256, 0, stream>>>(part, nrev, stats + 0);
  reduce_partial<<<NBLK_RED, 256, 0, stream>>>(item_rev, nrev, part);
  reduce_final<<<1, 256, 0, stream>>>(part, nrev, stats + 2);

  // P = norm(review) @ W + b   (FP32 WMMA, async-LDS staged A)
  gemm_norm_wmma<<<NU / 16, 256, 0, stream>>>(user_rev, stats + 0, W, bias, P_user);
  gemm_norm_wmma<<<NI / 16, 256, 0, stream>>>(item_rev, stats + 2, W, bias, P_item);

  // stats for the projected matrices
  reduce_partial<<<NBLK_RED, 256, 0, stream>>>(P_user, ED, part);
  reduce_final<<<1, 256, 0, stream>>>(part, ED, stats + 4);
  reduce_partial<<<NBLK_RED, 256, 0, stream>>>(P_item, ED, part);
  reduce_final<<<1, 256, 0, stream>>>(part, ED, stats + 6);

  // fusion_user / final_item
  fuse_kernel<<<8192, 256, 0, stream>>>(user_emb, item_emb, P_user, P_item,
                                        stats + 4, fusion_user, final_item);

  // zero the three spmm accumulators
  zero_kernel<<<8192, 256, 0, stream>>>(u_soc, 3 * ED);

  // three sparse propagations
  spmm_atomic<<<(NS + 7) / 8, 256, 0, stream>>>(soc_idx, soc_val, NS,
                                                fusion_user, u_soc);
  spmm_atomic<<<(NC + 7) / 8, 256, 0, stream>>>(con_idx, con_val, NC,
                                                final_item, u_items);
  spmm_atomic<<<(NS + 7) / 8, 256, 0, stream>>>(soc_idx, soc_val, NS, u_soc,
                                                u_soc2);

  // gather + dot + sigmoid
  score_kernel<<<(NB + 7) / 8, 256, 0, stream>>>(user_ids, item_ids, u_soc,
                                                 u_items, u_soc2, final_item,
                                                 out);
}